// GCODLoss_395136991275
// MI455X (gfx1250) — compile-verified
//
#include <hip/hip_runtime.h>
#include <math.h>

typedef __attribute__((ext_vector_type(2))) float v2f;
typedef __attribute__((ext_vector_type(8))) float v8f;

#define NUM_GRAPHS 256
#define LAMBDA 0.01f
#define FEAT 64

// ---------------------------------------------------------------------------
// zero a float buffer
// ---------------------------------------------------------------------------
__global__ void zero_kernel(float* __restrict__ p, int n) {
    int i = blockIdx.x * blockDim.x + threadIdx.x;
    if (i < n) p[i] = 0.0f;
}

// ---------------------------------------------------------------------------
// degree: deg[row[e]] += 1
// ---------------------------------------------------------------------------
__global__ void deg_kernel(const int* __restrict__ row, float* __restrict__ deg, int E) {
    int i = blockIdx.x * blockDim.x + threadIdx.x;
    if (i < E) atomicAdd(&deg[row[i]], 1.0f);
}

// ---------------------------------------------------------------------------
// deg -> deg^{-1/2} in place (0 where deg == 0)
// ---------------------------------------------------------------------------
__global__ void dis_kernel(float* __restrict__ deg, int N) {
    int i = blockIdx.x * blockDim.x + threadIdx.x;
    if (i < N) {
        float d = deg[i];
        deg[i] = (d > 0.0f) ? rsqrtf(d) : 0.0f;
    }
}

// ---------------------------------------------------------------------------
// Dirichlet energy per edge, segment-summed into 256 graph bins.
// One wave32 handles 16 edges per iteration:
//   lane L: edge e = L&15, dim-half h = L>>4 (dims [32h, 32h+32))
//   16 x V_WMMA_F32_16X16X4_F32 with A == B == diff tile -> diag = sum(diff^2)
//   (f32 A-layout == f32 B-layout when B = A^T, so the same VGPR pair feeds
//    both operands). Two independent accumulator chains of 8 WMMAs each to
//   halve the D->C dependency depth; merged with one v8f add.
//   diag(i): i in 0..7 -> lane i, acc[i]; i in 8..15 -> lane i+16, acc[i-8]
// ---------------------------------------------------------------------------
__global__ void __launch_bounds__(256)
energy_kernel(const float* __restrict__ x,
              const int*   __restrict__ row_idx,
              const int*   __restrict__ col_idx,
              const int*   __restrict__ batch,
              const float* __restrict__ dis,
              float* __restrict__ g_energy,   // [NUM_GRAPHS]
              float* __restrict__ g_count,    // [NUM_GRAPHS]
              int E)
{
    __shared__ float s_energy[NUM_GRAPHS];
    __shared__ float s_count[NUM_GRAPHS];
    // blockDim.x == 256 == NUM_GRAPHS: exactly one slot per thread, no divergence
    s_energy[threadIdx.x] = 0.0f;
    s_count[threadIdx.x]  = 0.0f;
    __syncthreads();

    const int lane   = threadIdx.x & 31;
    const int e_loc  = lane & 15;        // edge within 16-edge group
    const int half   = lane >> 4;        // 0 or 1
    const int dbase  = half * 32;        // dim base for this lane

    const int waves_per_block = blockDim.x >> 5;
    const int gwave  = blockIdx.x * waves_per_block + (threadIdx.x >> 5);
    const int nwaves = gridDim.x * waves_per_block;
    const int ngroups = (E + 15) >> 4;

    const bool owner = (lane < 8) || (lane >= 24);
    const int  comp  = (lane < 8) ? lane : (lane - 24);

    for (int g = gwave; g < ngroups; g += nwaves) {
        int  e      = g * 16 + e_loc;
        bool evalid = (e < E);
        int  ec     = evalid ? e : (E - 1);
        int  r      = row_idx[ec];
        int  c      = col_idx[ec];

        const float4* xr = (const float4*)(x + (size_t)r * FEAT + dbase);
        const float4* xc = (const float4*)(x + (size_t)c * FEAT + dbase);

        // 32 dims per lane = 8 x b128 gathers per endpoint
        float4 dr[8];
        #pragma unroll
        for (int t = 0; t < 8; ++t) {
            float4 a = xr[t];
            float4 b = xc[t];
            dr[t].x = a.x - b.x;
            dr[t].y = a.y - b.y;
            dr[t].z = a.z - b.z;
            dr[t].w = a.w - b.w;
        }
        if (!evalid) {
            #pragma unroll
            for (int t = 0; t < 8; ++t) { dr[t].x = 0.f; dr[t].y = 0.f; dr[t].z = 0.f; dr[t].w = 0.f; }
        }

        // Two independent accumulation chains (8 WMMAs each).
        v8f acc0 = {};
        v8f acc1 = {};
        #pragma unroll
        for (int t = 0; t < 8; ++t) {
            v2f a0; a0.x = dr[t].x; a0.y = dr[t].y;
            acc0 = __builtin_amdgcn_wmma_f32_16x16x4_f32(
                false, a0, false, a0, (short)0, acc0, false, false);
            v2f a1; a1.x = dr[t].z; a1.y = dr[t].w;
            acc1 = __builtin_amdgcn_wmma_f32_16x16x4_f32(
                false, a1, false, a1, (short)0, acc1, false, false);
        }
        v8f acc = acc0 + acc1;

        // diagonal element (per-edge sum of squared diffs) lives in `comp`
        float sq = 0.0f;
        #pragma unroll
        for (int k = 0; k < 8; ++k)
            if (k == comp) sq = acc[k];

        if (owner && evalid) {
            float nrm = dis[r] * dis[c];
            int   seg = batch[r];
            atomicAdd(&s_energy[seg], nrm * sq);
            atomicAdd(&s_count[seg], 1.0f);
        }
    }

    __syncthreads();
    atomicAdd(&g_energy[threadIdx.x], s_energy[threadIdx.x]);
    atomicAdd(&g_count[threadIdx.x],  s_count[threadIdx.x]);
}

// ---------------------------------------------------------------------------
// CE over 256x10 logits + combine with smoothness. One block of 256 threads.
// ---------------------------------------------------------------------------
__global__ void finalize_kernel(const float* __restrict__ logits,
                                const int*   __restrict__ labels,
                                const float* __restrict__ g_energy,
                                const float* __restrict__ g_count,
                                float* __restrict__ out)
{
    __shared__ float s_ce[NUM_GRAPHS];
    __shared__ float s_sm[NUM_GRAPHS];
    int t = threadIdx.x;

    const float* lg = logits + t * 10;
    float m = lg[0];
    #pragma unroll
    for (int k = 1; k < 10; ++k) m = fmaxf(m, lg[k]);
    float s = 0.0f;
    #pragma unroll
    for (int k = 0; k < 10; ++k) s += expf(lg[k] - m);
    float lse = m + logf(s);
    int lab = labels[t];
    s_ce[t] = -(lg[lab] - lse);

    float cnt = g_count[t];
    if (cnt < 1.0f) cnt = 1.0f;
    s_sm[t] = g_energy[t] / cnt;
    __syncthreads();

    #pragma unroll
    for (int off = NUM_GRAPHS / 2; off > 0; off >>= 1) {
        if (t < off) {
            s_ce[t] += s_ce[t + off];
            s_sm[t] += s_sm[t + off];
        }
        __syncthreads();
    }
    if (t == 0)
        out[0] = s_ce[0] / (float)NUM_GRAPHS
               + LAMBDA * (s_sm[0] / (float)NUM_GRAPHS);
}

// ---------------------------------------------------------------------------
extern "C" void kernel_launch(void* const* d_in, const int* in_sizes, int n_in,
                              void* d_out, int out_size, void* d_ws, size_t ws_size,
                              hipStream_t stream)
{
    const float* logits     = (const float*)d_in[0];       // [256,10]
    const int*   labels     = (const int*)  d_in[1];       // [256]
    const float* x          = (const float*)d_in[2];       // [N,64]
    const int*   edge_index = (const int*)  d_in[3];       // [2,E]
    const int*   batch      = (const int*)  d_in[4];       // [N]

    const int E = in_sizes[3] / 2;
    const int N = in_sizes[4];
    const int* row = edge_index;
    const int* col = edge_index + E;

    float* ws       = (float*)d_ws;
    float* deg      = ws;                       // [N] -> becomes deg^{-1/2}
    float* g_energy = ws + N;                   // [NUM_GRAPHS]
    float* g_count  = g_energy + NUM_GRAPHS;    // [NUM_GRAPHS]

    const int zn = N + 2 * NUM_GRAPHS;
    zero_kernel<<<(zn + 255) / 256, 256, 0, stream>>>(ws, zn);
    deg_kernel<<<(E + 255) / 256, 256, 0, stream>>>(row, deg, E);
    dis_kernel<<<(N + 255) / 256, 256, 0, stream>>>(deg, N);
    energy_kernel<<<512, 256, 0, stream>>>(x, row, col, batch, deg,
                                           g_energy, g_count, E);
    finalize_kernel<<<1, NUM_GRAPHS, 0, stream>>>(logits, labels,
                                                  g_energy, g_count,
                                                  (float*)d_out);
}